// MLPRefineSimilarity_29703993819993
// MI455X (gfx1250) — compile-verified
//
#include <hip/hip_runtime.h>

// Problem constants (from reference): N=16384 nodes, E=524288 edges, HID=128.
#define NNODE 16384
#define HID   128
#define NEDGE 524288

typedef float v2f __attribute__((ext_vector_type(2)));
typedef float v4f __attribute__((ext_vector_type(4)));
typedef float v8f __attribute__((ext_vector_type(8)));

// ---------------------------------------------------------------------------
// Kernel 1: per-node dual dot products via V_WMMA_F32_16X16X4_F32.
//   s1[n] = dot(emb[n], W[0:128]);  s2[n] = dot(emb[n], W[128:256])
// One wave32 per 16-row tile; D = A(16x4 f32) x B(4x16 f32) + C, K-loop of 32.
// B has W[0:128] in column 0, W[128:256] in column 1, columns 2..15 zero —
// built BRANCHLESSLY: every lane loads from a valid W address and multiplies
// by a per-lane 0/1 scale, so the loop body is loads + mul + wmma (no exec
// divergence). A layout (ISA 7.12.2, 32-bit A 16x4): lanes 0-15 -> K={0,1},
// lanes 16-31 -> K={2,3}; B mirrors with N=lane&15. EXEC all-1s (full wave).
// ---------------------------------------------------------------------------
__global__ __launch_bounds__(32) void node_scores_wmma(
    const float* __restrict__ emb, const float* __restrict__ W,
    float* __restrict__ s1, float* __restrict__ s2)
{
    const int lane  = threadIdx.x;        // 0..31
    const int m0    = blockIdx.x * 16;    // tile row base
    const int row   = m0 + (lane & 15);
    const int khalf = lane >> 4;          // 0 -> K pair {0,1}; 1 -> K pair {2,3}
    const int n     = lane & 15;          // B/D column carried by this lane

    // Branchless B source: lane n<2 reads its real W column, others read
    // column 0 but scale it to zero.
    const float scale = (n < 2) ? 1.0f : 0.0f;
    const v2f* __restrict__ e2 = (const v2f*)(emb + (size_t)row * HID);
    const v2f* __restrict__ w2 = (const v2f*)(W + (n == 1 ? HID : 0));

    v8f acc = {};
    #pragma unroll 4
    for (int kb2 = 0; kb2 < HID / 2; kb2 += 2) {   // kb2 counts v2f pairs
        v2f a  = e2[kb2 + khalf];                  // A[row][k0], A[row][k0+1]
        v2f bw = w2[kb2 + khalf];                  // W column slice
        v2f bm;
        bm.x = bw.x * scale;
        bm.y = bw.y * scale;
        // 8 args: (neg_a, A, neg_b, B, c_mod, C, reuse_a, reuse_b)
        acc = __builtin_amdgcn_wmma_f32_16x16x4_f32(
            false, a, false, bm, (short)0, acc, false, false);
    }

    // D layout (ISA 7.12.2, 32-bit 16x16 C/D): lanes 0-15 -> M=v, N=lane;
    // lanes 16-31 -> M=v+8, N=lane-16. We need columns 0 (s1) and 1 (s2).
    if (n <= 1) {
        float* __restrict__ dst = (n == 0) ? s1 : s2;
        const int mbase = m0 + khalf * 8;
        #pragma unroll
        for (int v = 0; v < 8; ++v) dst[mbase + v] = acc[v];
    }
}

// ---------------------------------------------------------------------------
// Kernel 2: zero the 1 GiB dense output with wide non-temporal stores.
// 268M floats -> 67M 16-byte stores; NT hint since output >> 192 MB L2.
// This kernel IS the roofline: ~46 us at 23.3 TB/s.
// ---------------------------------------------------------------------------
__global__ __launch_bounds__(256) void zero_out_kernel(
    v4f* __restrict__ z, long long nvec)
{
    long long i      = (long long)blockIdx.x * blockDim.x + threadIdx.x;
    long long stride = (long long)gridDim.x * blockDim.x;
    v4f zv = {0.f, 0.f, 0.f, 0.f};
    for (; i < nvec; i += stride) {
        __builtin_nontemporal_store(zv, &z[i]);
    }
}

// ---------------------------------------------------------------------------
// Kernel 3: per-edge scatter-add.
//   z[i0*N + i1] += s1[i0] + s2[i1] + b   (atomic: duplicates must sum)
// s1/s2 are 64 KB each -> live in L2; traffic is indices (8 B/edge) + ~67 MB
// of atomic RMW line traffic — small next to the 1 GiB fill.
// ---------------------------------------------------------------------------
__global__ __launch_bounds__(256) void scatter_edges_kernel(
    const int* __restrict__ idx, const float* __restrict__ s1,
    const float* __restrict__ s2, const float* __restrict__ bias,
    float* __restrict__ z)
{
    const int e = blockIdx.x * blockDim.x + threadIdx.x;
    if (e >= NEDGE) return;
    const int i0 = idx[e];           // v_indices[0][e]
    const int i1 = idx[NEDGE + e];   // v_indices[1][e]
    const float t = s1[i0] + s2[i1] + bias[0];
    atomicAdd(&z[(size_t)i0 * NNODE + i1], t);   // global_atomic_add_f32
}

// ---------------------------------------------------------------------------
// Host-side launcher (graph-capture safe: only kernel launches on `stream`).
// Inputs (setup_inputs order): 0=embeddings f32[N*HID], 1=v_indices int[2*E],
// 2=W f32[2*HID], 3=b f32[1]. Output: f32[N*N]. Workspace: s1,s2 (128 KB).
// ---------------------------------------------------------------------------
extern "C" void kernel_launch(void* const* d_in, const int* in_sizes, int n_in,
                              void* d_out, int out_size, void* d_ws, size_t ws_size,
                              hipStream_t stream) {
    const float* emb  = (const float*)d_in[0];
    const int*   idx  = (const int*)d_in[1];
    const float* W    = (const float*)d_in[2];
    const float* bias = (const float*)d_in[3];
    float*       z    = (float*)d_out;

    float* s1 = (float*)d_ws;
    float* s2 = s1 + NNODE;

    // 1) per-node scores (1024 waves, trivial cost, carries the WMMA math)
    node_scores_wmma<<<NNODE / 16, 32, 0, stream>>>(emb, W, s1, s2);

    // 2) zero 1 GiB output (bandwidth-bound)
    const long long nvec = (long long)NNODE * (long long)NNODE / 4;
    zero_out_kernel<<<8192, 256, 0, stream>>>((v4f*)z, nvec);

    // 3) scatter E edges with f32 atomics
    scatter_edges_kernel<<<NEDGE / 256, 256, 0, stream>>>(idx, s1, s2, bias, z);
}